// PH_16466904613333
// MI455X (gfx1250) — compile-verified
//
#include <hip/hip_runtime.h>
#include <hip/hip_bf16.h>
#include <math.h>

// ---------------------------------------------------------------------------
// Problem dims (compile-time constants from the reference)
// ---------------------------------------------------------------------------
#define BB    2
#define T_H   1024
#define T_A   128
#define HDIM  512
#define ADIM  14
#define EPSF  1e-5f
#define NA    4            // action rows tiled per block in ptr pass
#define TCH   4            // t-chunks per (side, row-group)

typedef __attribute__((ext_vector_type(2))) float v2f;
typedef __attribute__((ext_vector_type(8))) float v8f;

// tanh-approx GeLU (jax.nn.gelu approximate=True), exp-based tanh that
// saturates correctly for large |u|:  tanh(u) = 1 - 2/(exp(2u)+1)
__device__ __forceinline__ float gelu_f(float x) {
    float u  = 0.7978845608028654f * x * (1.0f + 0.044715f * x * x);
    float e  = __expf(2.0f * u);
    float th = 1.0f - 2.0f / (e + 1.0f);
    return 0.5f * x * (1.0f + th);
}

// ---------------------------------------------------------------------------
// GEMM: Y[M,512] = X[M,512] @ W[512,512] + bias[512]
// One wave per 16x16 output tile, V_WMMA_F32_16X16X4_F32, K steps of 4.
// Fragment layouts per CDNA5 ISA 7.12.2 (f32, 16x16x4).
// ---------------------------------------------------------------------------
__global__ void __launch_bounds__(32) gemm512_wmma(
    const float* __restrict__ X, const float* __restrict__ W,
    const float* __restrict__ bias, float* __restrict__ Y)
{
    const int tileM = blockIdx.x * 16;
    const int tileN = blockIdx.y * 16;
    const int lane  = threadIdx.x;      // 0..31
    const int half  = lane >> 4;        // 0 or 1
    const int mr    = lane & 15;

    v8f acc = {};
    const float* xrow = X + (size_t)(tileM + mr) * HDIM;

    #pragma unroll 4
    for (int k0 = 0; k0 < HDIM; k0 += 4) {
        const int kk = k0 + half * 2;
        v2f a, b;
        a.x = xrow[kk];
        a.y = xrow[kk + 1];
        b.x = W[(size_t)kk       * HDIM + tileN + mr];
        b.y = W[(size_t)(kk + 1) * HDIM + tileN + mr];
        acc = __builtin_amdgcn_wmma_f32_16x16x4_f32(
                  false, a, false, b, (short)0, acc, false, false);
    }

    const float bn = bias[tileN + mr];
    #pragma unroll
    for (int v = 0; v < 8; ++v) {
        int m = tileM + v + 8 * half;
        Y[(size_t)m * HDIM + tileN + mr] = acc[v] + bn;
    }
}

// ---------------------------------------------------------------------------
// Pointer head logits, fused + a-tiled:
//   logit[b,a,t] = inv*(s3 - m*c1) + c2, with
//   s1=sum g, s2=sum g^2, s3=sum g*(ln_g*w), g=gelu(H[b,t,h]+Q[b,a,h])
//   c1=sum ln_g*w, c2=sum ln_b*w + b_proj
// Each block: NA=4 consecutive action rows x one 256-wide t-chunk, so every
// H[t,:] tile fetched from L2 is reused for 4 logits (4x traffic reduction).
// gridDim.z selects left/right so both sides fill the machine in one launch.
// Logits go to workspace; log_softmax over t is a separate cheap pass.
// ---------------------------------------------------------------------------
__global__ void __launch_bounds__(256) ptr_logits_kernel(
    const float* __restrict__ Hl, const float* __restrict__ Ql,
    const float* __restrict__ lg, const float* __restrict__ lb,
    const float* __restrict__ lw, const float* __restrict__ lbp,
    const float* __restrict__ Hr, const float* __restrict__ Qr,
    const float* __restrict__ rg, const float* __restrict__ rb,
    const float* __restrict__ rw, const float* __restrict__ rbp,
    float* __restrict__ logits)   // [2, BB*T_A, T_H]
{
    __shared__ float q_s[NA * HDIM];
    __shared__ float gw_s[HDIM];
    __shared__ float red_s[256];

    const int side = blockIdx.z;
    const float* Hmat   = side ? Hr  : Hl;
    const float* Qmat   = side ? Qr  : Ql;
    const float* ln_g   = side ? rg  : lg;
    const float* ln_b   = side ? rb  : lb;
    const float* w_proj = side ? rw  : lw;
    const float* b_proj = side ? rbp : lbp;

    const int rowBase = blockIdx.x * NA;        // b*T_A + a (4 consecutive)
    const int b       = rowBase >> 7;           // T_A == 128, NA-aligned groups
    const int t0      = blockIdx.y * (T_H / TCH);
    const int tid     = threadIdx.x;
    const int lane    = tid & 31;
    const int wave    = tid >> 5;

    // Stage 4 q rows + ln_g*w into LDS; partial sums for c1, c2.
    float p1 = 0.0f, p2 = 0.0f;
    for (int h = tid; h < HDIM; h += 256) {
        float g = ln_g[h], w = w_proj[h];
        gw_s[h] = g * w;
        p1 += g * w;
        p2 += ln_b[h] * w;
    }
    for (int idx = tid; idx < NA * HDIM; idx += 256)
        q_s[idx] = Qmat[(size_t)rowBase * HDIM + idx];

    red_s[tid] = p1;
    __syncthreads();
    for (int s = 128; s > 0; s >>= 1) { if (tid < s) red_s[tid] += red_s[tid + s]; __syncthreads(); }
    const float c1 = red_s[0];
    __syncthreads();
    red_s[tid] = p2;
    __syncthreads();
    for (int s = 128; s > 0; s >>= 1) { if (tid < s) red_s[tid] += red_s[tid + s]; __syncthreads(); }
    const float c2 = red_s[0] + b_proj[0];
    __syncthreads();

    // Per-lane register copies of this lane's H-chunk of q (x4 rows) and gw.
    const int hbase = lane * 16;
    float qv[NA][16], gwv[16];
    #pragma unroll
    for (int j = 0; j < 16; ++j) gwv[j] = gw_s[hbase + j];
    #pragma unroll
    for (int a = 0; a < NA; ++a)
        #pragma unroll
        for (int j = 0; j < 16; ++j)
            qv[a][j] = q_s[a * HDIM + hbase + j];

    const float inv_h = 1.0f / (float)HDIM;
    for (int i = 0; i < (T_H / TCH) / 8; ++i) {
        const int t = t0 + wave + 8 * i;
        const float4* hp = (const float4*)(Hmat + ((size_t)(b * T_H + t)) * HDIM + hbase);
        float s1[NA], s2[NA], s3[NA];
        #pragma unroll
        for (int a = 0; a < NA; ++a) { s1[a] = 0.0f; s2[a] = 0.0f; s3[a] = 0.0f; }

        #pragma unroll
        for (int j4 = 0; j4 < 4; ++j4) {
            float4 hv = hp[j4];
            float xs[4] = {hv.x, hv.y, hv.z, hv.w};
            #pragma unroll
            for (int k = 0; k < 4; ++k) {
                const int j = j4 * 4 + k;
                #pragma unroll
                for (int a = 0; a < NA; ++a) {
                    float g = gelu_f(xs[k] + qv[a][j]);
                    s1[a] += g;
                    s2[a] += g * g;
                    s3[a] += g * gwv[j];
                }
            }
        }
        // wave32 butterfly reduction of all 12 partials
        #pragma unroll
        for (int off = 16; off > 0; off >>= 1) {
            #pragma unroll
            for (int a = 0; a < NA; ++a) {
                s1[a] += __shfl_xor(s1[a], off, 32);
                s2[a] += __shfl_xor(s2[a], off, 32);
                s3[a] += __shfl_xor(s3[a], off, 32);
            }
        }
        if (lane == 0) {
            #pragma unroll
            for (int a = 0; a < NA; ++a) {
                float m   = s1[a] * inv_h;
                float var = s2[a] * inv_h - m * m;
                float inv = rsqrtf(var + EPSF);
                logits[((size_t)side * BB * T_A + rowBase + a) * T_H + t] =
                    inv * (s3[a] - m * c1) + c2;
            }
        }
    }
}

// ---------------------------------------------------------------------------
// Row-wise log_softmax over T_H: out[row,t] = x[row,t] - max - log(sum exp)
// ---------------------------------------------------------------------------
__global__ void __launch_bounds__(256) row_logsoftmax_kernel(
    const float* __restrict__ x, float* __restrict__ out)
{
    __shared__ float red_s[256];
    const int row = blockIdx.x;
    const int tid = threadIdx.x;
    const float* xr = x + (size_t)row * T_H;

    float mx = -3.402823466e38f;
    for (int t = tid; t < T_H; t += 256) mx = fmaxf(mx, xr[t]);
    red_s[tid] = mx;
    __syncthreads();
    for (int s = 128; s > 0; s >>= 1) { if (tid < s) red_s[tid] = fmaxf(red_s[tid], red_s[tid + s]); __syncthreads(); }
    mx = red_s[0];
    __syncthreads();
    float se = 0.0f;
    for (int t = tid; t < T_H; t += 256) se += __expf(xr[t] - mx);
    red_s[tid] = se;
    __syncthreads();
    for (int s = 128; s > 0; s >>= 1) { if (tid < s) red_s[tid] += red_s[tid + s]; __syncthreads(); }
    const float lse = mx + logf(red_s[0]);
    __syncthreads();
    for (int t = tid; t < T_H; t += 256)
        out[(size_t)row * T_H + t] = xr[t] - lse;
}

// ---------------------------------------------------------------------------
// Action head tail: A1 (=heads@aW1+ab1) -> gelu -> LN -> @aW2+ab2 -> logsoftmax
// ---------------------------------------------------------------------------
__global__ void __launch_bounds__(256) action_kernel(
    const float* __restrict__ A1, const float* __restrict__ ln_g,
    const float* __restrict__ ln_b, const float* __restrict__ W2,
    const float* __restrict__ b2, float* __restrict__ out)
{
    __shared__ float red_s[256];
    __shared__ float logit_s[16];
    const int row = blockIdx.x;
    const int tid = threadIdx.x;

    float g0 = gelu_f(A1[(size_t)row * HDIM + tid]);
    float g1 = gelu_f(A1[(size_t)row * HDIM + tid + 256]);

    red_s[tid] = g0 + g1;
    __syncthreads();
    for (int s = 128; s > 0; s >>= 1) { if (tid < s) red_s[tid] += red_s[tid + s]; __syncthreads(); }
    const float m = red_s[0] * (1.0f / (float)HDIM);
    __syncthreads();
    red_s[tid] = g0 * g0 + g1 * g1;
    __syncthreads();
    for (int s = 128; s > 0; s >>= 1) { if (tid < s) red_s[tid] += red_s[tid + s]; __syncthreads(); }
    const float var = red_s[0] * (1.0f / (float)HDIM) - m * m;
    __syncthreads();
    const float inv = rsqrtf(var + EPSF);

    const float y0 = (g0 - m) * inv * ln_g[tid]       + ln_b[tid];
    const float y1 = (g1 - m) * inv * ln_g[tid + 256] + ln_b[tid + 256];

    for (int j = 0; j < ADIM; ++j) {
        red_s[tid] = y0 * W2[(size_t)tid * ADIM + j] + y1 * W2[(size_t)(tid + 256) * ADIM + j];
        __syncthreads();
        for (int s = 128; s > 0; s >>= 1) { if (tid < s) red_s[tid] += red_s[tid + s]; __syncthreads(); }
        if (tid == 0) logit_s[j] = red_s[0] + b2[j];
        __syncthreads();
    }

    if (tid == 0) {
        float mx = logit_s[0];
        for (int j = 1; j < ADIM; ++j) mx = fmaxf(mx, logit_s[j]);
        float se = 0.0f;
        for (int j = 0; j < ADIM; ++j) se += __expf(logit_s[j] - mx);
        logit_s[ADIM] = mx + logf(se);
    }
    __syncthreads();
    if (tid < ADIM)
        out[(size_t)row * ADIM + tid] = logit_s[tid] - logit_s[ADIM];
}

// ---------------------------------------------------------------------------
// Launch
// ---------------------------------------------------------------------------
extern "C" void kernel_launch(void* const* d_in, const int* in_sizes, int n_in,
                              void* d_out, int out_size, void* d_ws, size_t ws_size,
                              hipStream_t stream) {
    const float* hiddens = (const float*)d_in[0];   // [2,1024,512]
    const float* heads   = (const float*)d_in[1];   // [2,128,512]
    const float* aW1     = (const float*)d_in[2];
    const float* ab1     = (const float*)d_in[3];
    const float* aln_g   = (const float*)d_in[4];
    const float* aln_b   = (const float*)d_in[5];
    const float* aW2     = (const float*)d_in[6];
    const float* ab2     = (const float*)d_in[7];
    const float* lW_hid  = (const float*)d_in[8];
    const float* lb_hid  = (const float*)d_in[9];
    const float* lW_head = (const float*)d_in[10];
    const float* lb_head = (const float*)d_in[11];
    const float* lln_g   = (const float*)d_in[12];
    const float* lln_b   = (const float*)d_in[13];
    const float* lw_proj = (const float*)d_in[14];
    const float* lb_proj = (const float*)d_in[15];
    const float* rW_hid  = (const float*)d_in[16];
    const float* rb_hid  = (const float*)d_in[17];
    const float* rW_head = (const float*)d_in[18];
    const float* rb_head = (const float*)d_in[19];
    const float* rln_g   = (const float*)d_in[20];
    const float* rln_b   = (const float*)d_in[21];
    const float* rw_proj = (const float*)d_in[22];
    const float* rb_proj = (const float*)d_in[23];

    // Workspace layout (floats)
    float* ws = (float*)d_ws;
    float* Hl   = ws;                                 // 2048*512
    float* Hr   = Hl + (size_t)BB * T_H * HDIM;       // 2048*512
    float* Ql   = Hr + (size_t)BB * T_H * HDIM;       // 256*512
    float* Qr   = Ql + (size_t)BB * T_A * HDIM;       // 256*512
    float* A1   = Qr + (size_t)BB * T_A * HDIM;       // 256*512
    float* logi = A1 + (size_t)BB * T_A * HDIM;       // 2*256*1024

    float* out_actions = (float*)d_out;                            // 256*14
    float* out_lefts   = out_actions + (size_t)BB * T_A * ADIM;    // 256*1024
    float* out_rights  = out_lefts   + (size_t)BB * T_A * T_H;

    const dim3 wblk(32);
    const dim3 gridH((BB * T_H) / 16, HDIM / 16);    // 128 x 32
    const dim3 gridA((BB * T_A) / 16, HDIM / 16);    // 16 x 32

    gemm512_wmma<<<gridH, wblk, 0, stream>>>(hiddens, lW_hid, lb_hid, Hl);
    gemm512_wmma<<<gridH, wblk, 0, stream>>>(hiddens, rW_hid, rb_hid, Hr);
    gemm512_wmma<<<gridA, wblk, 0, stream>>>(heads, lW_head, lb_head, Ql);
    gemm512_wmma<<<gridA, wblk, 0, stream>>>(heads, rW_head, rb_head, Qr);
    gemm512_wmma<<<gridA, wblk, 0, stream>>>(heads, aW1, ab1, A1);

    action_kernel<<<BB * T_A, 256, 0, stream>>>(A1, aln_g, aln_b, aW2, ab2, out_actions);

    const dim3 pgrid((BB * T_A) / NA, TCH, 2);       // 64 x 4 x 2 = 512 blocks
    ptr_logits_kernel<<<pgrid, 256, 0, stream>>>(
        Hl, Ql, lln_g, lln_b, lw_proj, lb_proj,
        Hr, Qr, rln_g, rln_b, rw_proj, rb_proj, logi);

    row_logsoftmax_kernel<<<BB * T_A, 256, 0, stream>>>(logi, out_lefts);
    row_logsoftmax_kernel<<<BB * T_A, 256, 0, stream>>>(
        logi + (size_t)BB * T_A * T_H, out_rights);
}